// Atomic_HGNN_15522011808195
// MI455X (gfx1250) — compile-verified
//
#include <hip/hip_runtime.h>
#include <math.h>

// ---------------------------------------------------------------------------
// Problem constants (from the reference)
// ---------------------------------------------------------------------------
#define B_    1024
#define T_    5
#define N_    6
#define D_    192
#define E_    12
#define M_    12
#define L_    512
#define O_    6
#define SLICES (B_ * T_)        // 5120 (b,t) graphs
#define NODES  (SLICES * N_)    // 30720
#define XE     (SLICES * N_*N_) // 184320 edges
#define G3D    (3 * D_)         // 576 GRU rows

typedef float v2f __attribute__((ext_vector_type(2)));
typedef float v8f __attribute__((ext_vector_type(8)));

// Pair-interleaved activation layout: element (c, x) of a [C][X] matrix lives
// at  buf[((c>>1)*X + x)*2 + (c&1)].  A WMMA B-operand chunk (K rows kk,kk+1
// at column x) is then ONE aligned 8-byte load:  &buf[((2k+hi)*X + x)*2].

static __device__ __forceinline__ v8f wmma_f32(v2f a, v2f b, v8f c) {
  return __builtin_amdgcn_wmma_f32_16x16x4_f32(
      false, a, false, b, (short)0, c, false, false);
}
static __device__ __forceinline__ v8f vzero8() {
  v8f z = {0.f, 0.f, 0.f, 0.f, 0.f, 0.f, 0.f, 0.f};
  return z;
}
static __device__ __forceinline__ float sigm(float x) {
  return 1.0f / (1.0f + __expf(-x));
}
static __device__ __forceinline__ v2f ld2(const float* p) {
  return *reinterpret_cast<const v2f*>(p);
}
static __device__ __forceinline__ void st2(float* p, v2f v) {
  *reinterpret_cast<v2f*>(p) = v;
}

// ---------------------------------------------------------------------------
// K0: masked edge features -> pair-interleaved e2
// ---------------------------------------------------------------------------
__global__ void k_build_e(const float* __restrict__ pos,
                          const int* __restrict__ attmat,
                          float* __restrict__ e2) {
  int g = blockIdx.x * blockDim.x + threadIdx.x;
  if (g >= XE) return;
  int s = g / 36, idx = g % 36, i = idx / 6, j = idx % 6;
  float msk = ((attmat[g] == 1) && (i != j)) ? 1.f : 0.f;
  const float* pi = pos + (size_t)(s * 6 + i) * 6;
  const float* pj = pos + (size_t)(s * 6 + j) * 6;
  float v[12];
#pragma unroll
  for (int c = 0; c < 6; ++c) { v[c] = pi[c] * msk; v[c + 6] = pj[c] * msk; }
#pragma unroll
  for (int p = 0; p < 6; ++p) {
    v2f w = {v[2 * p], v[2 * p + 1]};
    st2(&e2[((size_t)p * XE + g) * 2], w);
  }
}

// ---------------------------------------------------------------------------
// K0b: node_resnet [B,T,D,N] -> h2 pair-interleaved [96][NODES][2]
// ---------------------------------------------------------------------------
__global__ void k_init_h(const float* __restrict__ nr, float* __restrict__ h2) {
  int q = blockIdx.x * blockDim.x + threadIdx.x;
  if (q >= (D_ / 2) * NODES) return;
  int p = q / NODES, node = q % NODES;
  int s = node / N_, n = node % N_;
  v2f w;
  w.x = nr[((size_t)s * D_ + 2 * p) * N_ + n];
  w.y = nr[((size_t)s * D_ + 2 * p + 1) * N_ + n];
  st2(&h2[(size_t)q * 2], w);
}

// ---------------------------------------------------------------------------
// KA: mh = Wh[12x192] @ h.  One wave per 32-node pair of tiles; the A operand
// (Wh row chunk) is loaded once and reused by two B tiles.
// ---------------------------------------------------------------------------
__global__ void k_mh(const float* __restrict__ Wh, const float* __restrict__ h2,
                     float* __restrict__ mh2) {
  int gwave = (blockIdx.x * blockDim.x + threadIdx.x) >> 5;
  int lane = threadIdx.x & 31;
  int hi = lane >> 4, l16 = lane & 15;
  int nodeBase = gwave * 32;
  // A rows >= 12 only affect D rows >= 12 which are never stored: clamp.
  int rowA = l16 < M_ ? l16 : (M_ - 1);
  const float* wrow  = Wh + (size_t)rowA * D_ + hi * 2;
  const float* bcol0 = h2 + ((size_t)hi * NODES + nodeBase + l16) * 2;
  const float* bcol1 = bcol0 + 32;  // +16 nodes
  v8f c0 = vzero8(), c1 = vzero8();
#pragma unroll 6
  for (int k = 0; k < D_ / 4; ++k) {
    v2f a = ld2(wrow + k * 4);
    c0 = wmma_f32(a, ld2(bcol0 + (size_t)k * 4 * NODES), c0);
    c1 = wmma_f32(a, ld2(bcol1 + (size_t)k * 4 * NODES), c1);
  }
#pragma unroll
  for (int pv = 0; pv < 4; ++pv) {
    int pair = pv + 4 * hi;
    if (pair < M_ / 2) {
      v2f w0 = {c0[2 * pv], c0[2 * pv + 1]};
      v2f w1 = {c1[2 * pv], c1[2 * pv + 1]};
      st2(&mh2[((size_t)pair * NODES + nodeBase + l16) * 2], w0);
      st2(&mh2[((size_t)pair * NODES + nodeBase + 16 + l16) * 2], w1);
    }
  }
}

// ---------------------------------------------------------------------------
// KB: per-(b,t) edge kernel, one wave per graph (36 edges -> 3 tiles of 16)
// ---------------------------------------------------------------------------
__global__ void k_edge(const float* __restrict__ es2,
                       const float* __restrict__ e2,
                       const float* __restrict__ W1, const float* __restrict__ b1,
                       const float* __restrict__ w2, const float* __restrict__ b2p,
                       const float* __restrict__ We, const float* __restrict__ mb,
                       const float* __restrict__ mh2, float* __restrict__ m2) {
  __shared__ __align__(16) float sW1[L_ * E_];   // 24 KB
  __shared__ __align__(16) v2f   sBW[L_];        // {b1,w2} interleaved, 4 KB
  __shared__ __align__(16) float sWe[M_ * E_];
  __shared__ float sMb[M_];
  __shared__ float sMh[8][M_ * N_];

  int tid = threadIdx.x;
  for (int q = tid; q < L_ * E_; q += 256) sW1[q] = W1[q];
  for (int q = tid; q < L_; q += 256) { v2f t = {b1[q], w2[q]}; sBW[q] = t; }
  for (int q = tid; q < M_ * E_; q += 256) sWe[q] = We[q];
  if (tid < M_) sMb[tid] = mb[tid];
  int sliceBase = blockIdx.x * 8;
  for (int q = tid; q < 8 * M_ * N_; q += 256) {
    int w = q / (M_ * N_), r = q % (M_ * N_);
    int c = r / N_, j = r % N_;
    int node = (sliceBase + w) * N_ + j;
    sMh[w][r] = mh2[(((size_t)(c >> 1) * NODES + node) * 2) + (c & 1)];
  }
  __syncthreads();

  int wave = tid >> 5, lane = tid & 31;
  int hi = lane >> 4, l16 = lane & 15;
  size_t gBase = (size_t)(sliceBase + wave) * 36;
  float b2 = b2p[0];
  int rowWe = l16 < M_ ? l16 : (M_ - 1);   // clamp (rows >=12 unused)

  for (int t = 0; t < 3; ++t) {
    int idx = t * 16 + l16;
    bool valid = idx < 36;
    int idxc = valid ? idx : 35;           // clamp: garbage cols never stored
    size_t g = gBase + idx;
    size_t gc = gBase + idxc;

    v2f bes[3], be[3];
#pragma unroll
    for (int k = 0; k < 3; ++k) {          // pair index 2k+hi
      size_t off = (((size_t)(2 * k + hi)) * XE + gc) * 2;
      bes[k] = ld2(&es2[off]);
      be[k]  = ld2(&e2[off]);
    }

    // ---- link: 32 row tiles of W1, relu + w2-weighted reduction ----
    // B operand (bes) lives in registers across the whole loop.
    float acc = 0.f;
#pragma unroll 2
    for (int rt = 0; rt < 32; ++rt) {
      v8f c = vzero8();
#pragma unroll
      for (int k = 0; k < 3; ++k) {
        v2f a = ld2(&sW1[(rt * 16 + l16) * E_ + k * 4 + hi * 2]);
        c = wmma_f32(a, bes[k], c);
      }
#pragma unroll
      for (int v = 0; v < 8; ++v) {
        v2f bw = sBW[rt * 16 + v + 8 * hi];
        float hm = c[v] + bw.x;
        hm = hm > 0.f ? hm : 0.f;
        acc += hm * bw.y;
      }
    }
    acc += __shfl_xor(acc, 16, 32);        // combine lane-halves (same edge)
    float adjv = sigm(acc + b2);

    // ---- me = We @ e ----
    v8f c2 = vzero8();
#pragma unroll
    for (int k = 0; k < 3; ++k) {
      v2f a = ld2(&sWe[rowWe * E_ + k * 4 + hi * 2]);
      c2 = wmma_f32(a, be[k], c2);
    }

    // ---- gated message, pair-interleaved store ----
    int j = idx % N_;
#pragma unroll
    for (int pv = 0; pv < 4; ++pv) {
      int pair = pv + 4 * hi;
      if (pair < M_ / 2 && valid) {
        int r0 = 2 * pair, r1 = r0 + 1;
        float m0 = sMh[wave][r0 * N_ + j] + c2[2 * pv] + sMb[r0];
        float m1 = sMh[wave][r1 * N_ + j] + c2[2 * pv + 1] + sMb[r1];
        m0 = m0 > 0.f ? m0 : 0.f;
        m1 = m1 > 0.f ? m1 : 0.f;
        v2f w = {adjv * m0, adjv * m1};
        st2(&m2[((size_t)pair * XE + g) * 2], w);
      }
    }
  }
}

// ---------------------------------------------------------------------------
// KC: msum pair (p, node) = sum_j m2[p][slice*36 + i*6 + j]
// ---------------------------------------------------------------------------
__global__ void k_msum(const float* __restrict__ m2, float* __restrict__ msum2) {
  int q = blockIdx.x * blockDim.x + threadIdx.x;
  if (q >= (M_ / 2) * NODES) return;
  int p = q / NODES, node = q % NODES;
  int s = node / N_, i = node % N_;
  size_t gb = (size_t)s * 36 + (size_t)i * N_;
  v2f acc = {0.f, 0.f};
#pragma unroll
  for (int j = 0; j < N_; ++j) acc += ld2(&m2[((size_t)p * XE + gb + j) * 2]);
  st2(&msum2[(size_t)q * 2], acc);
}

// ---------------------------------------------------------------------------
// KD: fused GRU. One workgroup (4 waves) per 16-node tile.  Each wave covers
// 9 row tiles in 3 groups of 3: three independent WMMA accumulator chains
// share every B-operand load (1.33 loads/WMMA, no C-dependency stalls).
// dynamic LDS: 2 * 576 * 16 floats = 73728 bytes
// ---------------------------------------------------------------------------
__global__ void k_gru(const float* __restrict__ Whh, const float* __restrict__ Wih,
                      const float* __restrict__ bih, const float* __restrict__ bhh,
                      const float* __restrict__ msum2, float* __restrict__ h2) {
  extern __shared__ float lds[];
  float* gi = lds;             // [576][16]
  float* gh = lds + G3D * 16;  // [576][16]

  int tid = threadIdx.x;
  int wave = tid >> 5, lane = tid & 31;
  int hi = lane >> 4, l16 = lane & 15;
  int nodeBase = blockIdx.x * 16;
  const float* bcolH = h2 + ((size_t)hi * NODES + nodeBase + l16) * 2;
  const float* bcolM = msum2 + ((size_t)hi * NODES + nodeBase + l16) * 2;

  for (int grp = 0; grp < 3; ++grp) {
    int rt0 = wave * 9 + grp * 3;
    const float* aH = Whh + (size_t)(rt0 * 16 + l16) * D_ + hi * 2;
    const float* aI = Wih + (size_t)(rt0 * 16 + l16) * M_ + hi * 2;
    v8f ch[3] = {vzero8(), vzero8(), vzero8()};
    v8f ci[3] = {vzero8(), vzero8(), vzero8()};
#pragma unroll 4
    for (int k = 0; k < D_ / 4; ++k) {      // Whh @ h, shared B chunk
      v2f b = ld2(bcolH + (size_t)k * 4 * NODES);
#pragma unroll
      for (int q = 0; q < 3; ++q)
        ch[q] = wmma_f32(ld2(aH + (size_t)q * 16 * D_ + k * 4), b, ch[q]);
    }
#pragma unroll
    for (int k = 0; k < 3; ++k) {           // Wih @ msum, shared B chunk
      v2f b = ld2(bcolM + (size_t)k * 4 * NODES);
#pragma unroll
      for (int q = 0; q < 3; ++q)
        ci[q] = wmma_f32(ld2(aI + (size_t)q * 16 * M_ + k * 4), b, ci[q]);
    }
#pragma unroll
    for (int q = 0; q < 3; ++q) {
#pragma unroll
      for (int v = 0; v < 8; ++v) {
        int row = (rt0 + q) * 16 + v + 8 * hi;
        gh[row * 16 + l16] = ch[q][v];
        gi[row * 16 + l16] = ci[q][v];
      }
    }
  }
  __syncthreads();

  for (int q = tid; q < (D_ / 2) * 16; q += 128) {
    int pp = q >> 4, col = q & 15;
    int node = nodeBase + col;
    int d0 = 2 * pp;
    v2f bir = ld2(&bih[d0]),        bhr = ld2(&bhh[d0]);
    v2f biz = ld2(&bih[d0 + D_]),   bhz = ld2(&bhh[d0 + D_]);
    v2f bin = ld2(&bih[d0 + 2*D_]), bhn = ld2(&bhh[d0 + 2*D_]);
    v2f hold = ld2(&h2[((size_t)pp * NODES + node) * 2]);
    v2f hnew;
#pragma unroll
    for (int e = 0; e < 2; ++e) {
      int d = d0 + e;
      float ir  = gi[d * 16 + col]            + bir[e];
      float hr  = gh[d * 16 + col]            + bhr[e];
      float iz  = gi[(d + D_) * 16 + col]     + biz[e];
      float hz  = gh[(d + D_) * 16 + col]     + bhz[e];
      float in_ = gi[(d + 2 * D_) * 16 + col] + bin[e];
      float hn  = gh[(d + 2 * D_) * 16 + col] + bhn[e];
      float r = sigm(ir + hr);
      float z = sigm(iz + hz);
      float n = tanhf(in_ + r * hn);
      hnew[e] = (1.f - z) * n + z * hold[e];
    }
    st2(&h2[((size_t)pp * NODES + node) * 2], hnew);
  }
}

// ---------------------------------------------------------------------------
// KE: out[b,o] = relu(sum_{d,t,n} h * conv1_w + b1), then 6x6 conv2
// ---------------------------------------------------------------------------
__global__ void k_out(const float* __restrict__ h2, const float* __restrict__ w1,
                      const float* __restrict__ c1b, const float* __restrict__ w2,
                      const float* __restrict__ c2b, float* __restrict__ out) {
  __shared__ float red[O_][64];
  __shared__ float mid[O_];
  int b = blockIdx.x, tid = threadIdx.x;
  float acc[O_] = {0.f, 0.f, 0.f, 0.f, 0.f, 0.f};
  for (int p = tid; p < D_ * T_ * N_; p += 64) {
    int d = p / (T_ * N_), r = p % (T_ * N_);
    int t = r / N_, n = r % N_;
    int node = (b * T_ + t) * N_ + n;
    float hv = h2[(((size_t)(d >> 1) * NODES + node) * 2) + (d & 1)];
#pragma unroll
    for (int o = 0; o < O_; ++o)
      acc[o] += hv * w1[(((size_t)o * D_ + d) * T_ + t) * N_ + n];
  }
#pragma unroll
  for (int o = 0; o < O_; ++o) red[o][tid] = acc[o];
  __syncthreads();
  if (tid < O_) {
    float s = 0.f;
    for (int k = 0; k < 64; ++k) s += red[tid][k];
    s += c1b[tid];
    mid[tid] = s > 0.f ? s : 0.f;
  }
  __syncthreads();
  if (tid < O_) {
    float s = c2b[tid];
#pragma unroll
    for (int k = 0; k < O_; ++k) s += w2[tid * O_ + k] * mid[k];
    out[(size_t)b * O_ + tid] = s;
  }
}

// ---------------------------------------------------------------------------
// Host side
// ---------------------------------------------------------------------------
extern "C" void kernel_launch(void* const* d_in, const int* in_sizes, int n_in,
                              void* d_out, int out_size, void* d_ws, size_t ws_size,
                              hipStream_t stream) {
  (void)in_sizes; (void)n_in; (void)out_size; (void)ws_size;

  const float* node_resnet = (const float*)d_in[0];
  const float* pos         = (const float*)d_in[1];
  const int*   attmat      = (const int*)d_in[2];
  const float* link_w1     = (const float*)d_in[3];
  const float* link_b1     = (const float*)d_in[4];
  const float* link_w2     = (const float*)d_in[5];
  const float* link_b2     = (const float*)d_in[6];
  const float* msg_Wh      = (const float*)d_in[7];
  const float* msg_We      = (const float*)d_in[8];
  const float* msg_b       = (const float*)d_in[9];
  const float* gru_Wih     = (const float*)d_in[10];
  const float* gru_Whh     = (const float*)d_in[11];
  const float* gru_bih     = (const float*)d_in[12];
  const float* gru_bhh     = (const float*)d_in[13];
  const float* conv1_w     = (const float*)d_in[14];
  const float* conv1_b     = (const float*)d_in[15];
  const float* conv2_w     = (const float*)d_in[16];
  const float* conv2_b     = (const float*)d_in[17];

  // workspace (floats): e2[6][XE][2], m2[6][XE][2], h2[96][NODES][2],
  //                     mh2[6][NODES][2], msum2[6][NODES][2]   (~44 MB)
  float* ws      = (float*)d_ws;
  float* e2      = ws;
  float* m2      = e2 + (size_t)M_ * XE;
  float* h2      = m2 + (size_t)M_ * XE;
  float* mh2     = h2 + (size_t)D_ * NODES;
  float* msum2   = mh2 + (size_t)M_ * NODES;

  k_build_e<<<(XE + 255) / 256, 256, 0, stream>>>(pos, attmat, e2);
  k_init_h<<<((D_ / 2) * NODES + 255) / 256, 256, 0, stream>>>(node_resnet, h2);

  for (int layer = 0; layer < 2; ++layer) {
    const float* es_in = (layer == 0) ? e2 : m2;
    k_mh<<<NODES / 32 / 8, 256, 0, stream>>>(msg_Wh, h2, mh2);
    k_edge<<<SLICES / 8, 256, 0, stream>>>(es_in, e2, link_w1, link_b1,
                                           link_w2, link_b2, msg_We, msg_b,
                                           mh2, m2);
    k_msum<<<((M_ / 2) * NODES + 255) / 256, 256, 0, stream>>>(m2, msum2);
    k_gru<<<NODES / 16, 128, 2 * G3D * 16 * sizeof(float), stream>>>(
        gru_Whh, gru_Wih, gru_bih, gru_bhh, msum2, h2);
  }

  k_out<<<B_, 64, 0, stream>>>(h2, conv1_w, conv1_b, conv2_w, conv2_b,
                               (float*)d_out);
}